// Word2vecEmbeddingInput_90615220011778
// MI455X (gfx1250) — compile-verified
//
#include <hip/hip_runtime.h>
#include <math.h>

#define VOCAB   100000
#define EMB     256
#define NS      256
#define BATCH   16384

#define LDA     260          // padded row stride (floats) -> conflict-free column reads
#define WAVES   4
#define BLOCK   (WAVES * 32)
#define ROWS_PER_WAVE 16
#define ROWS_PER_BLOCK (WAVES * ROWS_PER_WAVE)
#define NBLOCKS (BATCH / ROWS_PER_BLOCK)   // 256

typedef __attribute__((ext_vector_type(2))) float v2f;
typedef __attribute__((ext_vector_type(8))) float v8f;

// log(NUM_SAMPLED * log_uniform_prob(id))  (hardware v_log_f32 path)
__device__ __forceinline__ float log_q(int id) {
    float fid = (float)id;
    float p = (__logf(fid + 2.0f) - __logf(fid + 1.0f)) * (1.0f / __logf((float)VOCAB + 1.0f));
    return __logf((float)NS * p);
}

// softplus tail via hardware exp/log: log(1 + e^{-|x|})
__device__ __forceinline__ float softplus_tail(float x) {
    return __logf(1.0f + __expf(-fabsf(x)));
}
__device__ __forceinline__ float xent_pos(float x) {  // label = 1
    return fmaxf(x, 0.0f) - x + softplus_tail(x);
}
__device__ __forceinline__ float xent_neg(float x) {  // label = 0
    return fmaxf(x, 0.0f) + softplus_tail(x);
}

// ---------------------------------------------------------------------------
// Kernel 1: gather sampled weight rows into workspace + per-sample bias term.
// grid = 256 blocks (one per sampled id), 64 threads.
// ---------------------------------------------------------------------------
__global__ void w2v_prep(const float* __restrict__ W,
                         const float* __restrict__ bvec,
                         const int*   __restrict__ sids,
                         float* __restrict__ Ws,
                         float* __restrict__ sbias)
{
    int s   = blockIdx.x;
    int sid = sids[s];
    const float4* src = (const float4*)(W + (size_t)sid * EMB);
    float4*       dst = (float4*)(Ws + (size_t)s * EMB);
    dst[threadIdx.x] = src[threadIdx.x];       // 64 threads x 16B = 256 floats
    if (threadIdx.x == 0)
        sbias[s] = bvec[sid] - log_q(sid);
}

// ---------------------------------------------------------------------------
// Kernel 2: fused embed-gather + true-logit dot + WMMA f32 GEMM + xent reduce.
// grid = 256 blocks x 128 threads (4 wave32). Each wave owns 16 batch rows.
// ---------------------------------------------------------------------------
__global__ __launch_bounds__(BLOCK)
void w2v_main(const float* __restrict__ emb,
              const float* __restrict__ W,
              const float* __restrict__ bvec,
              const int*   __restrict__ inputs,
              const int*   __restrict__ labels,
              const float* __restrict__ Ws,
              const float* __restrict__ sbias,
              float* __restrict__ out_embed,
              float* __restrict__ partials)
{
    extern __shared__ float smem[];
    const int lane = threadIdx.x & 31;
    const int wave = threadIdx.x >> 5;

    float* As   = smem + wave * (ROWS_PER_WAVE * LDA);        // 16x260 per wave
    float* Bs   = smem + WAVES * ROWS_PER_WAVE * LDA;         // 16x260 shared
    float* wsum = Bs + 16 * LDA;

    const int rowBase = blockIdx.x * ROWS_PER_BLOCK + wave * ROWS_PER_WAVE;

    // ---- Phase A: gather embed rows -> LDS (A tile) + global output -------
    for (int r = 0; r < ROWS_PER_WAVE; ++r) {
        int b   = rowBase + r;
        int tok = inputs[b];
        const float4* src = (const float4*)(emb + (size_t)tok * EMB);
        float4*       dst = (float4*)(out_embed + (size_t)b * EMB);
        float4 v0 = src[lane];
        float4 v1 = src[lane + 32];
        dst[lane]      = v0;                       // global_store_b128
        dst[lane + 32] = v1;
        *(float4*)(As + r * LDA + lane * 4)       = v0;   // ds_store_b128
        *(float4*)(As + r * LDA + 128 + lane * 4) = v1;
    }

    float lossAcc = 0.0f;

    // ---- Phase B: true logits (per-row f32 dot + wave reduction) ----------
    for (int r = 0; r < ROWS_PER_WAVE; ++r) {
        int b   = rowBase + r;
        int lab = labels[b];
        const float4* tw = (const float4*)(W + (size_t)lab * EMB);
        float4 e0 = *(const float4*)(As + r * LDA + lane * 4);
        float4 e1 = *(const float4*)(As + r * LDA + 128 + lane * 4);
        float4 w0 = tw[lane];
        float4 w1 = tw[lane + 32];
        float acc = e0.x * w0.x + e0.y * w0.y + e0.z * w0.z + e0.w * w0.w
                  + e1.x * w1.x + e1.y * w1.y + e1.z * w1.z + e1.w * w1.w;
        #pragma unroll
        for (int off = 16; off > 0; off >>= 1)
            acc += __shfl_xor(acc, off, 32);
        if (lane == 0) {
            float x = acc + bvec[lab] - log_q(lab);
            lossAcc += xent_pos(x);
        }
    }

    // ---- Phase C: sampled logits via V_WMMA_F32_16X16X4_F32 ---------------
    // A frag (16x4 f32): lanes 0-15 -> M=lane, VGPR v + 2*(lane>>4) -> K.
    // B frag (4x16 f32): N = lane&15, K = v + 2*(lane>>4).
    // C/D (16x16 f32):   VGPR v, group g=lane>>4 -> M = v + 8g, N = lane&15.
    const int n0 = lane & 15;
    const int g  = lane >> 4;
    const float* ap = As + (lane & 15) * LDA + 2 * g;   // row M, K-pair base
    const float* bp = Bs + n0 * LDA + 2 * g;            // col N, K-pair base

    for (int j = 0; j < 16; ++j) {
        __syncthreads();
        // stage B tile j as Bs[n][k] (row-major in K, padded stride LDA)
        {
            int n  = threadIdx.x >> 3;            // 0..15
            int k0 = (threadIdx.x & 7) * 32;      // 0..224
            const float4* src = (const float4*)(Ws + (size_t)(j * 16 + n) * EMB + k0);
            float4*       dst = (float4*)(Bs + n * LDA + k0);
            #pragma unroll
            for (int q = 0; q < 8; ++q) dst[q] = src[q];
        }
        __syncthreads();

        v8f c = {};
        #pragma unroll 8
        for (int kk = 0; kk < 64; ++kk) {
            v2f a = *(const v2f*)(ap + kk * 4);   // ds_load_2addr_b64 pairs
            v2f b = *(const v2f*)(bp + kk * 4);
            c = __builtin_amdgcn_wmma_f32_16x16x4_f32(
                    false, a, false, b, (short)0, c, false, false);
        }

        float nb = sbias[j * 16 + n0];            // includes -log(NS*q(s))
        #pragma unroll
        for (int v = 0; v < 8; ++v) {
            float x = c[v] + nb;
            lossAcc += xent_neg(x);
        }
    }

    // ---- deterministic wave -> block reduction ----------------------------
    #pragma unroll
    for (int off = 16; off > 0; off >>= 1)
        lossAcc += __shfl_xor(lossAcc, off, 32);
    if (lane == 0) wsum[wave] = lossAcc;
    __syncthreads();
    if (threadIdx.x == 0)
        partials[blockIdx.x] = (wsum[0] + wsum[1]) + (wsum[2] + wsum[3]);
}

// ---------------------------------------------------------------------------
// Kernel 3: deterministic final reduction -> mean.
// ---------------------------------------------------------------------------
__global__ void w2v_reduce(const float* __restrict__ partials,
                           float* __restrict__ outScalar)
{
    __shared__ float sh[NBLOCKS];
    sh[threadIdx.x] = partials[threadIdx.x];
    __syncthreads();
    for (int s = NBLOCKS / 2; s > 0; s >>= 1) {
        if ((int)threadIdx.x < s) sh[threadIdx.x] += sh[threadIdx.x + s];
        __syncthreads();
    }
    if (threadIdx.x == 0)
        *outScalar = sh[0] * (1.0f / (float)BATCH);
}

// ---------------------------------------------------------------------------
extern "C" void kernel_launch(void* const* d_in, const int* in_sizes, int n_in,
                              void* d_out, int out_size, void* d_ws, size_t ws_size,
                              hipStream_t stream) {
    const float* embeddings = (const float*)d_in[0];   // (VOCAB, EMB)
    const float* nce_w      = (const float*)d_in[1];   // (VOCAB, EMB)
    const float* nce_b      = (const float*)d_in[2];   // (VOCAB,)
    const int*   inputs     = (const int*)  d_in[3];   // (BATCH,)
    const int*   labels     = (const int*)  d_in[4];   // (BATCH,1) flat
    const int*   sids       = (const int*)  d_in[5];   // (NS,)

    float* out = (float*)d_out;                        // [BATCH*EMB embed][1 cost]

    // workspace layout
    float* Ws       = (float*)d_ws;                    // NS*EMB gathered rows
    float* sbias    = Ws + NS * EMB;                   // NS
    float* partials = sbias + NS;                      // NBLOCKS

    w2v_prep<<<NS, 64, 0, stream>>>(nce_w, nce_b, sids, Ws, sbias);

    size_t smemBytes = (size_t)(WAVES * ROWS_PER_WAVE * LDA + 16 * LDA + 8) * sizeof(float);
    w2v_main<<<NBLOCKS, BLOCK, smemBytes, stream>>>(
        embeddings, nce_w, nce_b, inputs, labels, Ws, sbias, out, partials);

    w2v_reduce<<<1, NBLOCKS, 0, stream>>>(partials, out + (size_t)BATCH * EMB);
}